// TpnTaskLoss_Non_Param_33337536151702
// MI455X (gfx1250) — compile-verified
//
#include <hip/hip_runtime.h>
#include <math.h>

typedef __attribute__((ext_vector_type(2))) float v2f;
typedef __attribute__((ext_vector_type(8))) float v8f;

static constexpr int C    = 64;
static constexpr int DIM  = 1024;
static constexpr int NS   = 8192;
static constexpr int NT   = 8192;
static constexpr int NU   = 16384;
static constexpr int NTOT = NS + NT + NU;   // 32768

// workspace layout (in floats)
static constexpr size_t OFF_SUM_S = 0;
static constexpr size_t OFF_SUM_T = OFF_SUM_S + (size_t)C * DIM;
static constexpr size_t OFF_CNT_S = OFF_SUM_T + (size_t)C * DIM;
static constexpr size_t OFF_CNT_T = OFF_CNT_S + C;
static constexpr size_t OFF_U     = OFF_CNT_T + C;            // u_s, u_t, u_st contiguous
static constexpr size_t OFF_ACC   = OFF_U + 3ull * C * DIM;   // [kl, ce_s, ce_t, mmd]
static constexpr size_t WS_FLOATS = OFF_ACC + 4;

// ---------------------------------------------------------------- utilities
__global__ void zero_kernel(float* p, int n) {
    int i = blockIdx.x * blockDim.x + threadIdx.x;
    if (i < n) p[i] = 0.0f;
}

// ------------------------------------------------------------- label counts
__global__ void count_kernel(const int* __restrict__ slab,
                             const int* __restrict__ tlab,
                             float* __restrict__ cnt_s,
                             float* __restrict__ cnt_t) {
    __shared__ int cnt[C];
    const int* lab = blockIdx.x ? tlab : slab;
    float* out     = blockIdx.x ? cnt_t : cnt_s;
    int n          = blockIdx.x ? NT : NS;
    int tid = threadIdx.x;
    if (tid < C) cnt[tid] = 0;
    __syncthreads();
    for (int i = tid; i < n; i += blockDim.x) atomicAdd(&cnt[lab[i]], 1);
    __syncthreads();
    if (tid < C) out[tid] = (float)cnt[tid];
}

// ------------------------------------------------------------- segment sums
// grid (DIM/128 chunks, 16 sample splits, 2 matrices), 128 threads.
// Each thread exclusively owns one feature column in the LDS per-class acc,
// so the hot loop is one coalesced global load + one LDS fma per sample.
__global__ void segsum_kernel(const float* __restrict__ src,
                              const float* __restrict__ trg,
                              const int* __restrict__ slab,
                              const int* __restrict__ tlab,
                              float* __restrict__ sum_s,
                              float* __restrict__ sum_t) {
    __shared__ float acc[C * 128];
    const int chunk = blockIdx.x;          // 0..7  (128 dims each)
    const int split = blockIdx.y;          // 0..15 (512 samples each)
    const int mat   = blockIdx.z;
    const float* feat = mat ? trg : src;
    const int*   lab  = mat ? tlab : slab;
    float*       sum  = mat ? sum_t : sum_s;
    const int tid = threadIdx.x;

    for (int i = tid; i < C * 128; i += 128) acc[i] = 0.0f;
    __syncthreads();

    const int i0 = split * (NS / 16);
    const int i1 = i0 + (NS / 16);
    const int dbase = chunk * 128;
    for (int i = i0; i < i1; ++i) {
        int c = lab[i];                           // uniform -> scalar load
        acc[c * 128 + tid] += feat[(size_t)i * DIM + dbase + tid];
    }
    __syncthreads();
    for (int c = 0; c < C; ++c)
        atomicAdd(&sum[(size_t)c * DIM + dbase + tid], acc[c * 128 + tid]);
}

// --------------------------------------------------- prototypes + MMD loss
__global__ void proto_kernel(float* __restrict__ ws) {
    const float* sum_s = ws + OFF_SUM_S;
    const float* sum_t = ws + OFF_SUM_T;
    const float* cnt_s = ws + OFF_CNT_S;
    const float* cnt_t = ws + OFF_CNT_T;
    float* u_s  = ws + OFF_U;
    float* u_t  = u_s + (size_t)C * DIM;
    float* u_st = u_t + (size_t)C * DIM;
    float* accum = ws + OFF_ACC;

    __shared__ float red[256];
    const int c   = blockIdx.x;   // class
    const int tid = threadIdx.x;

    const float cs  = fmaxf(cnt_s[c], 1.0f);
    const float ct  = fmaxf(cnt_t[c], 1.0f);
    const float cst = fmaxf(cnt_s[c] + cnt_t[c], 1.0f);

    float local = 0.0f;
    for (int d = tid; d < DIM; d += 256) {
        size_t idx = (size_t)c * DIM + d;
        float ss = sum_s[idx], st = sum_t[idx];
        float us  = ss / cs;
        float ut  = st / ct;
        float ust = (ss + st) / cst;
        u_s[idx] = us; u_t[idx] = ut; u_st[idx] = ust;
        float d0 = us - ut, d1 = us - ust, d2 = ut - ust;
        local += d0 * d0 + d1 * d1 + d2 * d2;
    }
    red[tid] = local;
    __syncthreads();
    for (int s = 128; s > 0; s >>= 1) {
        if (tid < s) red[tid] += red[tid + s];
        __syncthreads();
    }
    if (tid == 0) atomicAdd(&accum[3], red[0]);
}

// ------------------------------------------- fused GEMM + softmax + CE + KL
static __device__ inline v8f wmma_f32(v2f a, v2f b, v8f c) {
    // V_WMMA_F32_16X16X4_F32 : D = A(16x4) * B(4x16) + C(16x16)
    return __builtin_amdgcn_wmma_f32_16x16x4_f32(
        false, a, false, b, (short)0, c, false, false);
}

static constexpr int KC   = 32;        // K-chunk staged in LDS
static constexpr int LDSS = 36;        // padded stride: 16B-aligned quads AND
                                       // conflict-free (4*(9l mod 16) distinct)

__global__ void __launch_bounds__(256)
gemm_loss_kernel(const float* __restrict__ src,
                 const float* __restrict__ trg,
                 const float* __restrict__ un,
                 const int* __restrict__ slab,
                 const int* __restrict__ tlab,
                 const float* __restrict__ u_base,   // u_s|u_t|u_st
                 float* __restrict__ accum) {
    __shared__ float sF[128 * LDSS];           // 128 rows x KC (padded)
    __shared__ float sU[3 * C * LDSS];         // 3 matrices x 64 classes x KC

    const int tid  = threadIdx.x;
    const int wv   = tid >> 5;
    const int lane = tid & 31;
    const int hf   = lane >> 4;    // half of wave
    const int grp  = lane & 15;

    const int rowbase = blockIdx.x * 128;      // region-aligned (128 | 8192)
    const float* fp; int ro;
    if (rowbase < NS)            { fp = src; ro = rowbase; }
    else if (rowbase < NS + NT)  { fp = trg; ro = rowbase - NS; }
    else                         { fp = un;  ro = rowbase - NS - NT; }

    v8f acc[3][4];
    const v8f vzero = {};
#pragma unroll
    for (int m = 0; m < 3; ++m)
#pragma unroll
        for (int t = 0; t < 4; ++t) acc[m][t] = vzero;

    for (int kc = 0; kc < DIM; kc += KC) {
        __syncthreads();
        // stage feature tile: 128x32 floats as b128 loads/stores
        for (int q = tid; q < 128 * KC / 4; q += 256) {
            int r  = q >> 3;                 // 8 quads per row
            int kk = (q & 7) << 2;
            float4 v = *reinterpret_cast<const float4*>(
                &fp[(size_t)(ro + r) * DIM + kc + kk]);
            *reinterpret_cast<float4*>(&sF[r * LDSS + kk]) = v;
        }
        // stage all three prototype matrices for this K-chunk
        for (int q = tid; q < 3 * C * KC / 4; q += 256) {
            int idx4 = q << 2;
            int m    = idx4 >> 11;           // / (C*KC)
            int rem  = idx4 & (C * KC - 1);
            int c    = rem >> 5;
            int kk   = rem & (KC - 1);
            float4 v = *reinterpret_cast<const float4*>(
                &u_base[(size_t)m * C * DIM + (size_t)c * DIM + kc + kk]);
            *reinterpret_cast<float4*>(&sU[m * C * LDSS + c * LDSS + kk]) = v;
        }
        // prefetch next K-chunk toward the WGP while this chunk computes
        if (kc + KC < DIM) {
            if (tid < 128)
                __builtin_prefetch(&fp[(size_t)(ro + tid) * DIM + kc + KC], 0, 3);
            else if (tid < 128 + 3 * C / 4) {
                int r = (tid - 128) * 4;     // 192 u-rows prefetched 4-at-a-time? no: row r
                __builtin_prefetch(&u_base[(size_t)r * DIM + kc + KC], 0, 3);
            }
        }
        __syncthreads();

#pragma unroll
        for (int k0 = 0; k0 < KC; k0 += 4) {
            // A frag: lanes 0-15 hold {K=k0,k0+1}, lanes 16-31 {k0+2,k0+3}
            v2f a;
            const int arow = (wv * 16 + grp) * LDSS + k0 + 2 * hf;
            a.x = sF[arow];
            a.y = sF[arow + 1];
            // gather ALL 12 B fragments first so the scheduler can batch the
            // dual LDS loads and wait with partial dscnt instead of 0
            v2f b[3][4];
#pragma unroll
            for (int m = 0; m < 3; ++m) {
#pragma unroll
                for (int t = 0; t < 4; ++t) {
                    const int bidx = m * C * LDSS + (t * 16 + grp) * LDSS + k0 + hf;
                    b[m][t].x = sU[bidx];
                    b[m][t].y = sU[bidx + 2];
                }
            }
#pragma unroll
            for (int m = 0; m < 3; ++m)
#pragma unroll
                for (int t = 0; t < 4; ++t)
                    acc[m][t] = wmma_f32(a, b[m][t], acc[m][t]);
        }
    }

    // ---- fused losses on register tiles ----------------------------------
    // acc[m][t][v]: row = v + 8*hf (within wave's 16-row tile), col = t*16+grp
    float klsum = 0.0f, cesum = 0.0f;
    const int wrow = rowbase + wv * 16;

#pragma unroll
    for (int v = 0; v < 8; ++v) {
        float lp[3][4];
#pragma unroll
        for (int m = 0; m < 3; ++m) {
            float x0 = acc[m][0][v], x1 = acc[m][1][v];
            float x2 = acc[m][2][v], x3 = acc[m][3][v];
            float mx = fmaxf(fmaxf(x0, x1), fmaxf(x2, x3));
#pragma unroll
            for (int off = 1; off < 16; off <<= 1)
                mx = fmaxf(mx, __shfl_xor(mx, off, 32));
            float se = __expf(x0 - mx) + __expf(x1 - mx) +
                       __expf(x2 - mx) + __expf(x3 - mx);
#pragma unroll
            for (int off = 1; off < 16; off <<= 1)
                se += __shfl_xor(se, off, 32);
            float lse = mx + __logf(se);
            lp[m][0] = x0 - lse; lp[m][1] = x1 - lse;
            lp[m][2] = x2 - lse; lp[m][3] = x3 - lse;
        }
        // symmetric-KL raw sums over the three pairs
#pragma unroll
        for (int t = 0; t < 4; ++t) {
            float a = lp[0][t], b = lp[1][t], c2 = lp[2][t];
            klsum += __expf(b)  * (b  - a) + __expf(a) * (a - b);
            klsum += __expf(c2) * (c2 - a) + __expf(a) * (a - c2);
            klsum += __expf(c2) * (c2 - b) + __expf(b) * (b - c2);
        }
        // cross entropy (label load uniform within each 16-lane half)
        const int row = v + 8 * hf;
        const int g   = wrow + row;
        if (g < NS) {
            int lab = slab[g];
#pragma unroll
            for (int t = 0; t < 4; ++t)
                if (t * 16 + grp == lab) cesum -= lp[0][t];
        } else if (g < NS + NT) {
            int lab = tlab[g - NS];
#pragma unroll
            for (int t = 0; t < 4; ++t)
                if (t * 16 + grp == lab) cesum -= lp[1][t];
        }
    }

    // wave reduction + single atomic per wave
#pragma unroll
    for (int off = 1; off < 32; off <<= 1) {
        klsum += __shfl_xor(klsum, off, 32);
        cesum += __shfl_xor(cesum, off, 32);
    }
    if (lane == 0) {
        atomicAdd(&accum[0], klsum);
        if (wrow < NS)            atomicAdd(&accum[1], cesum);
        else if (wrow < NS + NT)  atomicAdd(&accum[2], cesum);
    }
}

// ------------------------------------------------------------------ finalize
__global__ void finalize_kernel(const float* __restrict__ accum,
                                float* __restrict__ out) {
    if (threadIdx.x == 0 && blockIdx.x == 0) {
        out[0] = accum[0] / (6.0f * (float)NTOT * (float)C);   // loss_tpn
        out[1] = accum[1] / (float)NS;                         // loss_supv_s
        out[2] = accum[2] / (float)NT;                         // loss_supv_t
        out[3] = accum[3] / (3.0f * (float)(C * DIM));         // mmd_loss
    }
}

// -------------------------------------------------------------------- launch
extern "C" void kernel_launch(void* const* d_in, const int* in_sizes, int n_in,
                              void* d_out, int out_size, void* d_ws, size_t ws_size,
                              hipStream_t stream) {
    const float* src  = (const float*)d_in[0];
    const float* trg  = (const float*)d_in[1];
    const float* un   = (const float*)d_in[2];
    const int*   slab = (const int*)d_in[3];
    const int*   tlab = (const int*)d_in[4];
    float* ws  = (float*)d_ws;
    float* out = (float*)d_out;

    const int nz = (int)WS_FLOATS;
    zero_kernel<<<(nz + 255) / 256, 256, 0, stream>>>(ws, nz);

    count_kernel<<<2, 256, 0, stream>>>(slab, tlab, ws + OFF_CNT_S, ws + OFF_CNT_T);

    segsum_kernel<<<dim3(DIM / 128, 16, 2), 128, 0, stream>>>(
        src, trg, slab, tlab, ws + OFF_SUM_S, ws + OFF_SUM_T);

    proto_kernel<<<C, 256, 0, stream>>>(ws);

    gemm_loss_kernel<<<NTOT / 128, 256, 0, stream>>>(
        src, trg, un, slab, tlab, ws + OFF_U, ws + OFF_ACC);

    finalize_kernel<<<1, 64, 0, stream>>>(ws + OFF_ACC, out);
}